// Spectral_Weight_9637906612594
// MI455X (gfx1250) — compile-verified
//
#include <hip/hip_runtime.h>

// ---------------- problem constants ----------------
#define Cc   128
#define MIDc 32
#define Kk   7
#define K2   49
#define Gg   8
#define Hh   56
#define Ww   56
#define Bb   8
#define HW   (Hh * Ww)          // 3136
#define Nn   (Bb * HW)          // 25088
#define EPSf 1e-5f

typedef float v2f __attribute__((ext_vector_type(2)));
typedef float v8f __attribute__((ext_vector_type(8)));

// D = A(16x4, f32) * B(4x16, f32) + C(16x16, f32), full-precision CDNA5 WMMA
__device__ __forceinline__ v8f wmma_f32(v2f a, v2f b, v8f c) {
  return __builtin_amdgcn_wmma_f32_16x16x4_f32(
      /*neg_a=*/false, a, /*neg_b=*/false, b,
      /*c_mod=*/(short)0, c, /*reuse_a=*/false, /*reuse_b=*/false);
}

// ---------------------------------------------------------------------------
// Kernel 1: H1[32 x N] = w1[32 x 128] * X[128 x N]
// grid 196 x 256 threads; each wave computes a 32x16 output tile.
// ---------------------------------------------------------------------------
__global__ void k_gemm1(const float* __restrict__ X,
                        const float* __restrict__ w1,
                        float* __restrict__ H1) {
  const int tid    = threadIdx.x;
  const int wv     = tid >> 5;
  const int l      = tid & 31;
  const int lane16 = l & 15;
  const int hi     = l >> 4;

  const int p0   = blockIdx.x * 128 + wv * 16;
  const int pcol = p0 + lane16;                 // this lane's pixel column
  const int xb   = pcol / HW;
  const int s    = pcol % HW;
  const int xcb  = xb * Cc * HW + s;            // base for channel-strided loads

  v8f acc0 = {}; v8f acc1 = {};
  #pragma unroll 4
  for (int kk = 0; kk < 32; ++kk) {
    const int k0 = kk * 4 + 2 * hi;             // channel pair this lane feeds
    v2f b;
    b.x = X[xcb + k0 * HW];
    b.y = X[xcb + (k0 + 1) * HW];
    v2f a0 = *(const v2f*)(w1 + lane16 * Cc + k0);          // rows 0..15
    v2f a1 = *(const v2f*)(w1 + (16 + lane16) * Cc + k0);   // rows 16..31
    acc0 = wmma_f32(a0, b, acc0);
    acc1 = wmma_f32(a1, b, acc1);
  }
  #pragma unroll
  for (int r = 0; r < 8; ++r) {
    H1[(r + 8 * hi) * Nn + pcol]        = acc0[r];
    H1[(16 + r + 8 * hi) * Nn + pcol]   = acc1[r];
  }
}

// ---------------------------------------------------------------------------
// Kernel 2/4: per-channel mean & invstd over N (block per channel, LDS tree)
// ---------------------------------------------------------------------------
__global__ void k_stats(const float* __restrict__ src,
                        float* __restrict__ mean,
                        float* __restrict__ invstd) {
  __shared__ float s1[256];
  __shared__ float s2[256];
  const int c = blockIdx.x;
  const int t = threadIdx.x;
  float a = 0.f, b = 0.f;
  for (int i = t; i < Nn; i += 256) {
    float x = src[c * Nn + i];
    a += x; b += x * x;
  }
  s1[t] = a; s2[t] = b;
  __syncthreads();
  for (int o = 128; o > 0; o >>= 1) {
    if (t < o) { s1[t] += s1[t + o]; s2[t] += s2[t + o]; }
    __syncthreads();
  }
  if (t == 0) {
    float m = s1[0] * (1.0f / Nn);
    float v = s2[0] * (1.0f / Nn) - m * m;
    mean[c]   = m;
    invstd[c] = rsqrtf(v + EPSf);
  }
}

// ---------------------------------------------------------------------------
// Kernel 3 (fused): per 32-pixel tile
//   (a) LDS-stage hn = relu(BN(H1))                [32 x 32]
//   (b) WK = w2*hn + bias2 via WMMA -> LDS         [392(400) x 32]  (never to HBM)
//   (c) involution: X1[c,px] = sum_k2 patch*WK     -> LDS [128 x 32]
//   (d) X2 = w11 * X1 via WMMA -> d_out (pre-BN)
// dynamic LDS = (1024 + 12800 + 4096) * 4 = 71680 B
// ---------------------------------------------------------------------------
__global__ void k_fused(const float* __restrict__ X,
                        const float* __restrict__ H1,
                        const float* __restrict__ mean1,
                        const float* __restrict__ invstd1,
                        const float* __restrict__ g1,
                        const float* __restrict__ b1,
                        const float* __restrict__ w2,
                        const float* __restrict__ bias2,
                        const float* __restrict__ w11,
                        float* __restrict__ X2) {
  extern __shared__ float lds[];
  float* hn  = lds;            // 32*32
  float* wkl = lds + 1024;     // 400*32
  float* x1l = lds + 13824;    // 128*32

  const int tid    = threadIdx.x;
  const int wv     = tid >> 5;
  const int l      = tid & 31;
  const int lane16 = l & 15;
  const int hi     = l >> 4;
  const int p0     = blockIdx.x * 32;

  // (a) stage normalized H1 tile (coalesced: consecutive tid -> consecutive px)
  #pragma unroll
  for (int j = 0; j < 4; ++j) {
    const int idx = tid + 256 * j;       // m*32 + px
    const int m   = idx >> 5;
    const int px  = idx & 31;
    float x = H1[m * Nn + p0 + px];
    float y = (x - mean1[m]) * invstd1[m] * g1[m] + b1[m];
    hn[idx] = fmaxf(y, 0.f);
  }
  __syncthreads();

  // (b) GEMM2: 25 m-tiles x 2 n-tiles, K=32 (8 wmma steps)
  for (int job = wv; job < 50; job += 8) {
    const int mt  = job >> 1;
    const int nt  = job & 1;
    const int px  = nt * 16 + lane16;
    const int row = mt * 16 + lane16;    // A row (pad 392 -> 400)
    v8f acc = {};
    #pragma unroll
    for (int kk = 0; kk < 8; ++kk) {
      const int k0 = kk * 4 + 2 * hi;
      v2f b; b.x = hn[k0 * 32 + px]; b.y = hn[(k0 + 1) * 32 + px];
      v2f a;
      if (row < K2 * Gg) a = *(const v2f*)(w2 + row * MIDc + k0);
      else { a.x = 0.f; a.y = 0.f; }
      acc = wmma_f32(a, b, acc);
    }
    #pragma unroll
    for (int r = 0; r < 8; ++r) {
      const int rr = mt * 16 + r + 8 * hi;
      if (rr < K2 * Gg) wkl[rr * 32 + px] = acc[r] + bias2[rr];
    }
  }
  __syncthreads();

  // (c) involution: 128 channels x 32 pixels = 4096 outputs, 16 per thread
  #pragma unroll
  for (int j = 0; j < 16; ++j) {
    const int lin = tid + 256 * j;
    const int px  = lin & 31;
    const int c   = lin >> 5;
    const int p   = p0 + px;
    const int b_  = p / HW;
    const int s   = p % HW;
    const int h   = s / Ww;
    const int w   = s % Ww;
    const float* xb  = X + (b_ * Cc + c) * HW;
    const float* wkp = wkl + (c >> 4) * (K2 * 32) + px;
    float acc = 0.f;
    #pragma unroll
    for (int kh = 0; kh < Kk; ++kh) {
      const int hh = h + kh - 3;
      if ((unsigned)hh < (unsigned)Hh) {
        const float* xr = xb + hh * Ww;
        #pragma unroll
        for (int kw = 0; kw < Kk; ++kw) {
          const int ww2 = w + kw - 3;
          if ((unsigned)ww2 < (unsigned)Ww)
            acc += xr[ww2] * wkp[(kh * Kk + kw) * 32];
        }
      }
    }
    x1l[c * 32 + px] = acc;
  }
  __syncthreads();

  // (d) GEMM3: X2 = w11[128x128] * X1[128x32]; 8 m-tiles x 2 n-tiles
  for (int job = wv; job < 16; job += 8) {
    const int mt  = job >> 1;
    const int nt  = job & 1;
    const int px  = nt * 16 + lane16;
    const int row = mt * 16 + lane16;
    v8f acc = {};
    #pragma unroll 4
    for (int kk = 0; kk < 32; ++kk) {
      const int k0 = kk * 4 + 2 * hi;
      v2f b; b.x = x1l[k0 * 32 + px]; b.y = x1l[(k0 + 1) * 32 + px];
      v2f a = *(const v2f*)(w11 + row * Cc + k0);
      acc = wmma_f32(a, b, acc);
    }
    #pragma unroll
    for (int r = 0; r < 8; ++r) {
      const int rr = mt * 16 + r + 8 * hi;
      X2[rr * Nn + p0 + px] = acc[r];
    }
  }
}

// ---------------------------------------------------------------------------
// Kernel 5: in-place BN + ReLU on X2 (= d_out)
// ---------------------------------------------------------------------------
__global__ void k_bnrelu(float* __restrict__ y,
                         const float* __restrict__ mean,
                         const float* __restrict__ invstd,
                         const float* __restrict__ gamma,
                         const float* __restrict__ beta) {
  const int i = blockIdx.x * 256 + threadIdx.x;   // exactly C*N threads
  const int c = i / Nn;
  float x = y[i];
  float v = (x - mean[c]) * invstd[c] * gamma[c] + beta[c];
  y[i] = fmaxf(v, 0.f);
}

// ---------------------------------------------------------------------------
extern "C" void kernel_launch(void* const* d_in, const int* in_sizes, int n_in,
                              void* d_out, int out_size, void* d_ws, size_t ws_size,
                              hipStream_t stream) {
  (void)in_sizes; (void)n_in; (void)out_size; (void)ws_size;
  const float* X       = (const float*)d_in[0];
  const float* w1      = (const float*)d_in[1];
  const float* g1      = (const float*)d_in[2];
  const float* b1      = (const float*)d_in[3];
  const float* w2      = (const float*)d_in[4];
  const float* bias2   = (const float*)d_in[5];
  const float* w11     = (const float*)d_in[6];
  const float* gamma_h = (const float*)d_in[7];
  const float* beta_h  = (const float*)d_in[8];
  float* out = (float*)d_out;

  // workspace layout (floats): H1[32*N] | mean1[32] | invstd1[32] | mean2[128] | invstd2[128]
  float* ws      = (float*)d_ws;
  float* H1      = ws;
  float* mean1   = ws + MIDc * Nn;
  float* invstd1 = mean1 + MIDc;
  float* mean2   = invstd1 + MIDc;
  float* invstd2 = mean2 + Cc;

  k_gemm1<<<Nn / 128, 256, 0, stream>>>(X, w1, H1);
  k_stats<<<MIDc, 256, 0, stream>>>(H1, mean1, invstd1);
  k_fused<<<Nn / 32, 256, 17920 * sizeof(float), stream>>>(
      X, H1, mean1, invstd1, g1, b1, w2, bias2, w11, out);
  k_stats<<<Cc, 256, 0, stream>>>(out, mean2, invstd2);
  k_bnrelu<<<(Cc * Nn) / 256, 256, 0, stream>>>(out, mean2, invstd2, gamma_h, beta_h);
}